// EnhancedCoconutWithGNN_21749714387571
// MI455X (gfx1250) — compile-verified
//
#include <hip/hip_runtime.h>

#define Nn   50000
#define Ee   400000
#define INF_ 256   // in_features
#define Hh   8
#define Dd   32
#define HD   256   // H*D
#define EFe  3

typedef __attribute__((ext_vector_type(2))) float v2f;
typedef __attribute__((ext_vector_type(8))) float v8f;

// ---------- order-preserving float<->uint mapping for atomic segment-max ----------
__device__ __forceinline__ unsigned f2o(float f) {
  unsigned u = __float_as_uint(f);
  return (u & 0x80000000u) ? ~u : (u | 0x80000000u);
}
__device__ __forceinline__ float o2f(unsigned u) {
  return (u & 0x80000000u) ? __uint_as_float(u & 0x7fffffffu) : __uint_as_float(~u);
}

// ---------- init: out = bias broadcast, denom = 0, amax = "-inf" ----------
__global__ void init_kernel(float* __restrict__ out, float* __restrict__ denom,
                            unsigned* __restrict__ amax_u, const float* __restrict__ bias) {
  int idx = blockIdx.x * blockDim.x + threadIdx.x;
  if (idx < Nn * HD) out[idx] = bias[idx & (HD - 1)];
  if (idx < Nn * Hh) { denom[idx] = 0.f; amax_u[idx] = 0u; }  // 0u < f2o(any float)
}

// ---------- h = x @ W^T via V_WMMA_F32_16X16X4_F32, LDS-tiled ----------
// Block tile: M=32 x N=128. 8 waves; wave w: mtile = w&1, ntiles {2*(w>>1), 2*(w>>1)+1}.
__global__ __launch_bounds__(256) void gemm_h_wmma(const float* __restrict__ x,
                                                   const float* __restrict__ Wm,
                                                   float* __restrict__ h) {
  __shared__ float sA[32][33];    // x chunk  [M=32][K=32], padded
  __shared__ float sB[128][33];   // W chunk  [N=128][K=32] (B = W^T)
  const int tid   = threadIdx.x;
  const int mBase = blockIdx.x * 32;
  const int nBase = blockIdx.y * 128;
  const int wave  = tid >> 5, ln = tid & 31;
  const int half  = ln >> 4, mr = ln & 15;   // lanes 16-31 handle K+2..3 / M+8
  const int mt    = wave & 1;
  const int nt0   = (wave >> 1) << 1;

  v8f acc0 = {};
  v8f acc1 = {};

  for (int k0 = 0; k0 < INF_; k0 += 32) {
    // cooperative loads, coalesced within rows
    for (int i = tid; i < 32 * 32; i += 256) {
      int r = i >> 5, c = i & 31, row = mBase + r;
      sA[r][c] = (row < Nn) ? x[row * INF_ + k0 + c] : 0.f;
    }
    for (int i = tid; i < 128 * 32; i += 256) {
      int r = i >> 5, c = i & 31;
      sB[r][c] = Wm[(nBase + r) * INF_ + k0 + c];
    }
    __syncthreads();

    const int ar  = mt * 16 + mr;
    const int br0 = nt0 * 16 + mr;
    const int br1 = br0 + 16;
#pragma unroll
    for (int kk = 0; kk < 32; kk += 4) {
      const int kh = kk + half * 2;          // lanes 0-15: K0/K1, lanes 16-31: K2/K3
      v2f a, b0, b1;
      a.x  = sA[ar][kh];   a.y  = sA[ar][kh + 1];
      b0.x = sB[br0][kh];  b0.y = sB[br0][kh + 1];
      b1.x = sB[br1][kh];  b1.y = sB[br1][kh + 1];
      acc0 = __builtin_amdgcn_wmma_f32_16x16x4_f32(false, a, false, b0, (short)0, acc0, false, false);
      acc1 = __builtin_amdgcn_wmma_f32_16x16x4_f32(false, a, false, b1, (short)0, acc1, false, false);
    }
    __syncthreads();
  }

  // C/D layout: VGPR r -> M = r (lanes 0-15) / r+8 (lanes 16-31), N = mr
  const int rowBase = mBase + mt * 16 + half * 8;
#pragma unroll
  for (int r = 0; r < 8; ++r) {
    int row = rowBase + r;
    if (row < Nn) {
      h[row * HD + nBase + nt0 * 16 + mr]       = acc0[r];
      h[row * HD + nBase + (nt0 + 1) * 16 + mr] = acc1[r];
    }
  }
}

// ---------- per-edge attention logits + segment max (wave32 per edge, lane = d) ----------
__global__ __launch_bounds__(256) void edge_alpha_kernel(
    const float* __restrict__ h, const float* __restrict__ a_src,
    const float* __restrict__ a_dst, const float* __restrict__ edge_W,
    const float* __restrict__ edge_b, const float* __restrict__ edge_attr,
    const int* __restrict__ ei, float* __restrict__ alpha,
    unsigned* __restrict__ amax_u) {
  int e = blockIdx.x * 8 + (threadIdx.x >> 5);
  if (e >= Ee) return;
  int lane = threadIdx.x & 31;
  int src = ei[e], dst = ei[Ee + e];
  float ea0 = edge_attr[e * EFe + 0];
  float ea1 = edge_attr[e * EFe + 1];
  float ea2 = edge_attr[e * EFe + 2];
  const float* hs_row = h + (long)src * HD;
  const float* hd_row = h + (long)dst * HD;

  for (int hh = 0; hh < Hh; ++hh) {
    int hd_i = hh * Dd + lane;
    float hs  = hs_row[hd_i];
    float hdd = hd_row[hd_i];
    float enc = fmaf(ea0, edge_W[hd_i * EFe + 0],
                fmaf(ea1, edge_W[hd_i * EFe + 1],
                fmaf(ea2, edge_W[hd_i * EFe + 2], edge_b[hd_i])));
    enc = fmaxf(enc, 0.f);
    float p = hs * a_src[hd_i] + hdd * a_dst[hd_i] + enc * hdd;
#pragma unroll
    for (int off = 16; off > 0; off >>= 1) p += __shfl_xor(p, off, 32);
    if (lane == 0) {
      float al = (p >= 0.f) ? p : 0.2f * p;        // LeakyReLU(0.2)
      alpha[e * Hh + hh] = al;
      atomicMax(&amax_u[dst * Hh + hh], f2o(al));
    }
  }
}

// ---------- ex = exp(alpha - amax[dst]); denom[dst] += ex ----------
__global__ void softmax_num_kernel(const int* __restrict__ ei,
                                   const unsigned* __restrict__ amax_u,
                                   float* __restrict__ alpha_ex,
                                   float* __restrict__ denom) {
  int idx = blockIdx.x * blockDim.x + threadIdx.x;
  if (idx >= Ee * Hh) return;
  int e = idx >> 3, hh = idx & 7;
  int dst = ei[Ee + e];
  float m  = o2f(amax_u[dst * Hh + hh]);
  float ex = __expf(alpha_ex[idx] - m);
  alpha_ex[idx] = ex;
  atomicAdd(&denom[dst * Hh + hh], ex);
}

// ---------- out[dst] += h[src] * attn (wave32 per edge) ----------
__global__ __launch_bounds__(256) void scatter_out_kernel(
    const float* __restrict__ h, const float* __restrict__ ex,
    const float* __restrict__ denom, const int* __restrict__ ei,
    float* __restrict__ out) {
  int e = blockIdx.x * 8 + (threadIdx.x >> 5);
  if (e >= Ee) return;
  int lane = threadIdx.x & 31;
  int src = ei[e], dst = ei[Ee + e];
  const float* hs_row = h + (long)src * HD;
  float* out_row = out + (long)dst * HD;
  for (int hh = 0; hh < Hh; ++hh) {
    float attn = ex[e * Hh + hh] / fmaxf(denom[dst * Hh + hh], 1e-10f);
    atomicAdd(&out_row[hh * Dd + lane], hs_row[hh * Dd + lane] * attn);
  }
}

extern "C" void kernel_launch(void* const* d_in, const int* in_sizes, int n_in,
                              void* d_out, int out_size, void* d_ws, size_t ws_size,
                              hipStream_t stream) {
  const float* x        = (const float*)d_in[0];  // [N,256]
  const float* Wm       = (const float*)d_in[1];  // [256,256]
  const float* a_src    = (const float*)d_in[2];  // [8,32]
  const float* a_dst    = (const float*)d_in[3];  // [8,32]
  const float* edge_W   = (const float*)d_in[4];  // [256,3]
  const float* edge_b   = (const float*)d_in[5];  // [256]
  const float* bias     = (const float*)d_in[6];  // [256]
  const float* edge_attr= (const float*)d_in[7];  // [E,3]
  const int*   ei       = (const int*)d_in[8];    // [2,E]
  float* out = (float*)d_out;

  // workspace layout (floats): h[N*HD] | alpha/ex[E*H] | amax_u[N*H] | denom[N*H]
  float*    h       = (float*)d_ws;
  float*    alpha   = h + (size_t)Nn * HD;
  unsigned* amax_u  = (unsigned*)(alpha + (size_t)Ee * Hh);
  float*    denom   = (float*)(amax_u + (size_t)Nn * Hh);

  init_kernel<<<(Nn * HD + 255) / 256, 256, 0, stream>>>(out, denom, amax_u, bias);

  dim3 g((Nn + 31) / 32, HD / 128);
  gemm_h_wmma<<<g, 256, 0, stream>>>(x, Wm, h);

  edge_alpha_kernel<<<(Ee + 7) / 8, 256, 0, stream>>>(h, a_src, a_dst, edge_W, edge_b,
                                                      edge_attr, ei, alpha, amax_u);
  softmax_num_kernel<<<(Ee * Hh + 255) / 256, 256, 0, stream>>>(ei, amax_u, alpha, denom);
  scatter_out_kernel<<<(Ee + 7) / 8, 256, 0, stream>>>(h, alpha, denom, ei, out);
}